// LSTM_76046690943148
// MI455X (gfx1250) — compile-verified
//
#include <hip/hip_runtime.h>

// LSTM: B=32, S=1024, I=512, H=512 on MI455X (gfx1250, wave32, WMMA + TDM).
//
// Persistent grid-resident kernel, 8 blocks x 256 thr (= 64 waves).
// Per timestep:
//   stage  : TDM (tensor_load_to_lds) copies this block's 16 rows of x_t and
//            h (2 x 32KB) into LDS; s_wait_tensorcnt + __syncthreads.
//   phase 1: gates[32,2048] = bias + x_t @ W_ih^T + h @ W_hh^T via
//            v_wmma_f32_16x16x4_f32; each wave owns a 16x64 gate tile
//            (4 independent accumulator chains -> no inter-WMMA delay slots);
//            A fragments from LDS (ds_load_2addr_b64), B fragments (weights)
//            streamed from L2 (8MB resident in 192MB L2).
//   phase 2: elementwise i/f/g/o sigmoid/tanh, update c,h, write outputs.
// Two device-wide barriers per step (monotonic atomic counter in d_ws).

#define BB 32
#define SS 1024
#define II 512
#define HH 512
#define GG (4 * HH)          // 2048 gate columns

#define NBLK 8
#define TPB  256
#define NTHR (NBLK * TPB)    // 2048 threads = 64 waves
#define SUBN 4               // 16x16 subtiles per wave along N (16x64 region)
#define SMEM_BYTES (2 * 16 * II * 4)   // x_t tile + h tile, 16 rows each

typedef __attribute__((ext_vector_type(2))) float v2f;
typedef __attribute__((ext_vector_type(8))) float v8f;
typedef unsigned int v4u __attribute__((ext_vector_type(4)));
typedef int v8i __attribute__((ext_vector_type(8)));
typedef int v4i __attribute__((ext_vector_type(4)));

__device__ __forceinline__ v8f wmma_f32_16x16x4(v2f a, v2f b, v8f c) {
  return __builtin_amdgcn_wmma_f32_16x16x4_f32(
      false, a, false, b, (short)0, c, false, false);
}

__device__ __forceinline__ float sigm(float x) {
  return 1.0f / (1.0f + __expf(-x));
}
__device__ __forceinline__ float tanh_fast(float x) {
  return 1.0f - 2.0f / (__expf(2.0f * x) + 1.0f);
}

__device__ __forceinline__ void grid_barrier(unsigned* cnt, unsigned target) {
  __threadfence();
  __syncthreads();
  if (threadIdx.x == 0) {
    __hip_atomic_fetch_add(cnt, 1u, __ATOMIC_ACQ_REL, __HIP_MEMORY_SCOPE_AGENT);
    while (__hip_atomic_load(cnt, __ATOMIC_ACQUIRE,
                             __HIP_MEMORY_SCOPE_AGENT) < target) {
      __builtin_amdgcn_s_sleep(4);
    }
  }
  __syncthreads();
}

// ---- TDM: issue a 2-D tile load (rows x 512 f32) global -> LDS ----------
// D# per CDNA5 ISA ch.8: group0 = {flags/count, lds_addr, gaddr_lo, gaddr_hi
// | type=2<<30}; group1 packs data_size=4B, tensor/tile dims, row stride.
__device__ __forceinline__ void tdm_load_tile(const float* gptr,
                                              unsigned lds_off,
                                              unsigned rows,
                                              unsigned long long row_stride) {
#if __has_builtin(__builtin_amdgcn_tensor_load_to_lds)
  unsigned long long ga = (unsigned long long)(uintptr_t)gptr;
  v4u g0 = {1u,                                 // count=1, user D#
            lds_off,                            // lds_addr (bytes)
            (unsigned)(ga & 0xFFFFFFFFu),
            (unsigned)((ga >> 32) & 0x1FFFFFFu) | (2u << 30)};  // type=2
  unsigned sd0 = 512u;                          // tensor_dim0 (f32 units)
  v8i g1 = {(int)(2u << 16),                    // data_size=2 (4 bytes)
            (int)((sd0 & 0xFFFFu) << 16),       // tensor_dim0[15:0]
            (int)((sd0 >> 16) | ((BB & 0xFFFFu) << 16)),  // dim0_hi|dim1_lo
            (int)((512u) << 16),                // tile_dim0 = 512
            (int)rows,                          // tile_dim1
            (int)(unsigned)(row_stride & 0xFFFFFFFFull),  // dim0_stride lo
            (int)(unsigned)(row_stride >> 32),            // dim0_stride hi
            0};
  v4i gz = {0, 0, 0, 0};
#if __clang_major__ >= 23
  v8i gz8 = {0, 0, 0, 0, 0, 0, 0, 0};
  __builtin_amdgcn_tensor_load_to_lds(g0, g1, gz, gz, gz8, 0);
#else
  __builtin_amdgcn_tensor_load_to_lds(g0, g1, gz, gz, 0);
#endif
#else
  (void)gptr; (void)lds_off; (void)rows; (void)row_stride;
#endif
}

__device__ __forceinline__ void tdm_wait() {
#if __has_builtin(__builtin_amdgcn_s_wait_tensorcnt)
  __builtin_amdgcn_s_wait_tensorcnt(0);
#endif
}

__global__ void __launch_bounds__(TPB) lstm_persistent(
    const float* __restrict__ x,    // [B,S,I]
    const float* __restrict__ h0,   // [1,B,H]
    const float* __restrict__ c0,   // [1,B,H]
    const float* __restrict__ Wih,  // [4H,I]
    const float* __restrict__ Whh,  // [4H,H]
    const float* __restrict__ bih,  // [4H]
    const float* __restrict__ bhh,  // [4H]
    float* __restrict__ out,        // [B,S,H]
    float* __restrict__ hn,         // [B,H]
    float* __restrict__ cn,         // [B,H]
    float* __restrict__ gates,      // ws: [B,4H]
    float* __restrict__ hbuf,       // ws: [B,H]
    float* __restrict__ cbuf,       // ws: [B,H]
    unsigned* __restrict__ barrier_cnt) {
  extern __shared__ float smem[];           // [16][512] x_t, then [16][512] h
  float* lds_x = smem;
  float* lds_h = smem + 16 * II;

  const int tid   = blockIdx.x * TPB + (int)threadIdx.x;  // 0..2047
  const int lane  = threadIdx.x & 31;
  const int wave  = tid >> 5;       // 0..63
  const int mTile = wave >> 5;      // 0..1 (all waves of a block share mTile)
  const int nGrp  = wave & 31;      // 0..31
  const int m0    = mTile * 16;
  const int n0    = nGrp * 64;
  const int l15   = lane & 15;
  const int khalf = (lane >> 4) * 2;

  // ---- init h, c from h0, c0 ----
  for (int e = tid; e < BB * HH; e += NTHR) {
    hbuf[e] = h0[e];
    cbuf[e] = c0[e];
  }
  unsigned gen = 0;
  grid_barrier(barrier_cnt, ++gen * NBLK);

  const float* wi[SUBN];
  const float* wh[SUBN];
  float bias[SUBN];
  int   col[SUBN];
#pragma unroll
  for (int s = 0; s < SUBN; ++s) {
    col[s]  = n0 + 16 * s + l15;
    wi[s]   = Wih + (size_t)col[s] * II;
    wh[s]   = Whh + (size_t)col[s] * HH;
    bias[s] = bih[col[s]] + bhh[col[s]];
  }

  const unsigned lds_x_off = (unsigned)(uintptr_t)lds_x;  // LDS byte offsets
  const unsigned lds_h_off = (unsigned)(uintptr_t)lds_h;
  const float* xtile0 = x + (size_t)m0 * (SS * II);  // rows m0..m0+15
  const float* htile0 = hbuf + (size_t)m0 * HH;

  for (int t = 0; t < SS; ++t) {
    // ---- stage A tiles into LDS via TDM (wave 0 only; EXEC ignored) ----
    if ((threadIdx.x >> 5) == 0) {
#if __has_builtin(__builtin_amdgcn_tensor_load_to_lds)
      tdm_load_tile(xtile0 + (size_t)t * II, lds_x_off, 16,
                    (unsigned long long)SS * II);
      tdm_load_tile(htile0, lds_h_off, 16, (unsigned long long)HH);
      tdm_wait();
#else
      // Fallback: cooperative copy by wave 0.
      for (int e = (int)threadIdx.x; e < 16 * II; e += 32) {
        lds_x[e] = xtile0[(size_t)(e >> 9) * (SS * II) + (size_t)t * II +
                          (e & (II - 1))];
        lds_h[e] = htile0[(size_t)(e >> 9) * HH + (e & (HH - 1))];
      }
#endif
    }
    __syncthreads();

    // ---- phase 1: gate GEMM (K = I + H = 1024), A from LDS ----
    v8f acc[SUBN];
#pragma unroll
    for (int s = 0; s < SUBN; ++s) {
#pragma unroll
      for (int r = 0; r < 8; ++r) acc[s][r] = bias[s];
    }

    const float* ax = lds_x + l15 * II + khalf;
    const float* ah = lds_h + l15 * HH + khalf;
#pragma unroll 4
    for (int k = 0; k < II; k += 4) {
      v2f a = *(const v2f*)(ax + k);
#pragma unroll
      for (int s = 0; s < SUBN; ++s) {
        v2f b = *(const v2f*)(wi[s] + k + khalf);
        acc[s] = wmma_f32_16x16x4(a, b, acc[s]);
      }
    }
#pragma unroll 4
    for (int k = 0; k < HH; k += 4) {
      v2f a = *(const v2f*)(ah + k);
#pragma unroll
      for (int s = 0; s < SUBN; ++s) {
        v2f b = *(const v2f*)(wh[s] + k + khalf);
        acc[s] = wmma_f32_16x16x4(a, b, acc[s]);
      }
    }

    // ---- store gate tile (C/D layout: VGPR r -> M = r + 8*(lane>>4)) ----
    const int rbase = m0 + (lane >> 4) * 8;
#pragma unroll
    for (int s = 0; s < SUBN; ++s) {
#pragma unroll
      for (int r = 0; r < 8; ++r) {
        gates[(rbase + r) * GG + col[s]] = acc[s][r];
      }
    }
    grid_barrier(barrier_cnt, ++gen * NBLK);

    // ---- phase 2: elementwise LSTM cell update ----
    for (int e = tid; e < BB * HH; e += NTHR) {
      const int b = e >> 9;
      const int j = e & (HH - 1);
      const float* grow = gates + b * GG;
      float ig = sigm(grow[j]);
      float fg = sigm(grow[j + HH]);
      float gg = tanh_fast(grow[j + 2 * HH]);
      float og = sigm(grow[j + 3 * HH]);
      float cv = fg * cbuf[e] + ig * gg;
      float hv = og * tanh_fast(cv);
      cbuf[e] = cv;
      hbuf[e] = hv;
      out[(size_t)b * (SS * HH) + (size_t)t * HH + j] = hv;
      if (t == SS - 1) {
        hn[e] = hv;
        cn[e] = cv;
      }
    }
    grid_barrier(barrier_cnt, ++gen * NBLK);
  }
}

extern "C" void kernel_launch(void* const* d_in, const int* in_sizes, int n_in,
                              void* d_out, int out_size, void* d_ws,
                              size_t ws_size, hipStream_t stream) {
  const float* x   = (const float*)d_in[0];
  const float* h0  = (const float*)d_in[1];
  const float* c0  = (const float*)d_in[2];
  const float* Wih = (const float*)d_in[3];
  const float* Whh = (const float*)d_in[4];
  const float* bih = (const float*)d_in[5];
  const float* bhh = (const float*)d_in[6];

  float* out = (float*)d_out;                 // [B,S,H]
  float* hn  = out + (size_t)BB * SS * HH;    // [1,B,H]
  float* cn  = hn + (size_t)BB * HH;          // [1,B,H]

  unsigned char* ws = (unsigned char*)d_ws;
  unsigned* cnt = (unsigned*)ws;                        // 256 B barrier region
  float* gates  = (float*)(ws + 256);                   // B*4H floats (256 KB)
  float* hbuf   = gates + (size_t)BB * GG;              // B*H  floats (64 KB)
  float* cbuf   = hbuf + (size_t)BB * HH;               // B*H  floats (64 KB)

  hipMemsetAsync(cnt, 0, 256, stream);
  lstm_persistent<<<NBLK, TPB, SMEM_BYTES, stream>>>(x, h0, c0, Wih, Whh, bih,
                                                     bhh, out, hn, cn, gates,
                                                     hbuf, cbuf, cnt);
}